// Net_19619410608498
// MI455X (gfx1250) — compile-verified
//
#include <hip/hip_runtime.h>
#include <hip/hip_bf16.h>
#include <cfloat>

typedef float v2f __attribute__((ext_vector_type(2)));
typedef float v8f __attribute__((ext_vector_type(8)));

#define N_ALL 8600
#define N_PMT 600
#define N_VOX 8000
#define NAP   8608   // N_ALL padded to multiple of 16

// ---------------- utility kernels ----------------

__global__ __launch_bounds__(256) void zero_i32(int* __restrict__ p, int n) {
    int i = blockIdx.x * 256 + threadIdx.x;
    if (i < n) p[i] = 0;
}

// Build padded node features: [NAP][64]; cols 0..50 = inputs (PMT rows) else 0,
// cols 51..53 = koor, cols 54..63 = 0, pad rows = 0.
__global__ __launch_bounds__(256) void build_x(const float* __restrict__ inp,
                                               const float* __restrict__ koor,
                                               float* __restrict__ x) {
    int i = blockIdx.x * 256 + threadIdx.x;
    if (i >= NAP * 64) return;
    int r = i >> 6, c = i & 63;
    float v = 0.f;
    if (r < N_ALL) {
        if (c < 51)      v = (r < N_PMT) ? inp[r * 51 + c] : 0.f;
        else if (c < 54) v = koor[r * 3 + (c - 51)];
    }
    x[i] = v;
}

// ---------------- CSR build (count -> scan -> fill) ----------------

__global__ __launch_bounds__(256) void count_edges(const int* __restrict__ dst, int nE,
                                                   int* __restrict__ cnt) {
    int e = blockIdx.x * 256 + threadIdx.x;
    if (e < nE) atomicAdd(&cnt[dst[e]], 1);
}

__global__ __launch_bounds__(256) void scan_counts(const int* __restrict__ cnt, int n, int total,
                                                   int* __restrict__ roff) {
    __shared__ int part[256];
    int t = threadIdx.x;
    int chunk = (n + 255) >> 8;
    int lo = t * chunk;
    int hi = (lo + chunk < n) ? (lo + chunk) : n;
    if (lo > n) lo = n;
    int s = 0;
    for (int i = lo; i < hi; ++i) s += cnt[i];
    part[t] = s;
    __syncthreads();
    if (t == 0) {
        int run = 0;
        for (int i = 0; i < 256; ++i) { int tmp = part[i]; part[i] = run; run += tmp; }
    }
    __syncthreads();
    int run = part[t];
    for (int i = lo; i < hi; ++i) { roff[i] = run; run += cnt[i]; }
    if (t == 0) roff[n] = total;
}

__global__ __launch_bounds__(256) void fill_csr(const int* __restrict__ src,
                                                const int* __restrict__ dst, int nE,
                                                const int* __restrict__ roff,
                                                int* __restrict__ cur,
                                                int* __restrict__ out) {
    int e = blockIdx.x * 256 + threadIdx.x;
    if (e < nE) {
        int d = dst[e];
        int pos = atomicAdd(&cur[d], 1);
        out[roff[d] + pos] = src[e];
    }
}

// ---------------- node GEMM: Y = X*Wt, P = X*Wp, f32 WMMA 16x16x4 ----------------
// X: [>=ntiles*16][KP] (zero-padded). Wt/Wp: [K][Nout] row-major from inputs.
// Y/P written [rows][NP], pad columns get exact zeros (weights zero-padded).
template <int KP, int NP>
__global__ __launch_bounds__(256) void gemm_yp(const float* __restrict__ X,
                                               const float* __restrict__ Wt,
                                               const float* __restrict__ Wp,
                                               int K, int Nout, int ntiles,
                                               float* __restrict__ Y, float* __restrict__ P) {
    constexpr int NPS = NP + 8;   // LDS row pad: keeps the two B reads on disjoint bank sets
    constexpr int NT  = NP / 16;  // output 16-col tiles
    __shared__ float wl[2 * KP * NPS];
    float* wlt = wl;
    float* wlp = wl + KP * NPS;

    // Stage zero-padded weights into LDS (cooperative, all 256 threads).
    for (int idx = threadIdx.x; idx < KP * NP; idx += 256) {
        int kk = idx / NP, oo = idx % NP;
        bool inb = (kk < K) && (oo < Nout);
        wlt[kk * NPS + oo] = inb ? Wt[kk * Nout + oo] : 0.f;
        wlp[kk * NPS + oo] = inb ? Wp[kk * Nout + oo] : 0.f;
    }
    __syncthreads();

    int rt = blockIdx.x * 8 + (threadIdx.x >> 5);  // one 16-row tile per wave (wave32)
    if (rt >= ntiles) return;                      // uniform per-wave, EXEC stays all-ones

    const int lane = threadIdx.x & 31;
    const int mrow = lane & 15;          // A: M index
    const int koff = (lane >> 4) << 1;   // A/B: lanes 16-31 hold K+2,K+3
    const int col  = lane & 15;          // B/C/D: N index

    v8f accY[NT] = {};
    v8f accP[NT] = {};

    const float* xrow = X + (size_t)(rt * 16 + mrow) * KP;
#pragma unroll
    for (int k = 0; k < KP; k += 4) {
        v2f a = *(const v2f*)(xrow + k + koff);   // A frag: K=k+koff, k+koff+1
#pragma unroll
        for (int t = 0; t < NT; ++t) {
            v2f b0;
            b0.x = wlt[(k + koff)     * NPS + t * 16 + col];
            b0.y = wlt[(k + koff + 1) * NPS + t * 16 + col];
            accY[t] = __builtin_amdgcn_wmma_f32_16x16x4_f32(false, a, false, b0,
                                                            (short)0, accY[t], false, false);
            v2f b1;
            b1.x = wlp[(k + koff)     * NPS + t * 16 + col];
            b1.y = wlp[(k + koff + 1) * NPS + t * 16 + col];
            accP[t] = __builtin_amdgcn_wmma_f32_16x16x4_f32(false, a, false, b1,
                                                            (short)0, accP[t], false, false);
        }
    }

    // C/D layout: VGPR g -> rows (g, g+8) for lane halves, col = lane&15.
    const int rbase = rt * 16 + ((lane >> 4) << 3);
#pragma unroll
    for (int t = 0; t < NT; ++t) {
#pragma unroll
        for (int g = 0; g < 8; ++g) {
            size_t o = (size_t)(rbase + g) * NP + t * 16 + col;
            Y[o] = accY[t][g];
            P[o] = accP[t][g];
        }
    }
}

// ---------------- edge aggregation + finalize (one wave per destination) ----------------
// h[d] = (max over in-neighbors of Y[src]) - Y[node] + bt + P[node] + bp, opt dropout/final ReLU.
template <int NP>
__global__ __launch_bounds__(256) void edge_fin(const float* __restrict__ Y,
                                                const float* __restrict__ P,
                                                const int* __restrict__ roff,
                                                const int* __restrict__ rsrc,
                                                const float* __restrict__ bt,
                                                const float* __restrict__ bp,
                                                int Kout, int ndst, int node_base,
                                                const float* __restrict__ u,
                                                float* __restrict__ out, int final1) {
    int d = blockIdx.x * 8 + (threadIdx.x >> 5);
    if (d >= ndst) return;
    const int lane = threadIdx.x & 31;
    const int f0 = (NP == 16) ? (lane & 15) : lane;
    const int node = node_base + d;
    const int s = roff[node], e = roff[node + 1];

    float m0 = -FLT_MAX, m1 = -FLT_MAX;
    for (int j = s; j < e; ++j) {
        const float* yr = Y + (size_t)rsrc[j] * NP;
        m0 = fmaxf(m0, yr[f0]);
        if (NP == 64) m1 = fmaxf(m1, yr[f0 + 32]);
    }

    const float* yn = Y + (size_t)node * NP;
    const float* pn = P + (size_t)node * NP;

    float v0 = m0 - yn[f0] + pn[f0] + ((f0 < Kout) ? (bt[f0] + bp[f0]) : 0.f);
    if (u && f0 < Kout) v0 *= (u[(size_t)d * Kout + f0] > 0.5f) ? 2.f : 0.f;

    if (final1) {
        if (lane == 0) out[d] = fmaxf(v0, 0.f);
        return;
    }
    out[(size_t)d * NP + f0] = v0;

    if (NP == 64) {
        const int f1 = f0 + 32;
        float v1 = m1 - yn[f1] + pn[f1] + ((f1 < Kout) ? (bt[f1] + bp[f1]) : 0.f);
        if (u && f1 < Kout) v1 *= (u[(size_t)d * Kout + f1] > 0.5f) ? 2.f : 0.f;
        out[(size_t)d * NP + f1] = v1;
    }
}

// ---------------- host orchestration ----------------

extern "C" void kernel_launch(void* const* d_in, const int* in_sizes, int n_in,
                              void* d_out, int out_size, void* d_ws, size_t ws_size,
                              hipStream_t stream) {
    (void)n_in; (void)out_size; (void)ws_size;

    const float* inputs = (const float*)d_in[0];
    const float* koor   = (const float*)d_in[1];
    const float* u1  = (const float*)d_in[2];
    const float* u2  = (const float*)d_in[3];
    const float* u3  = (const float*)d_in[4];
    const float* u4  = (const float*)d_in[5];
    const float* bwt = (const float*)d_in[6];
    const float* bbt = (const float*)d_in[7];
    const float* bwp = (const float*)d_in[8];
    const float* bbp = (const float*)d_in[9];
    const float* awt = (const float*)d_in[10];
    const float* abt = (const float*)d_in[11];
    const float* awp = (const float*)d_in[12];
    const float* abp = (const float*)d_in[13];
    const float* a5wt = (const float*)d_in[14];
    const float* a5bt = (const float*)d_in[15];
    const float* a5wp = (const float*)d_in[16];
    const float* a5bp = (const float*)d_in[17];
    const float* cwt = (const float*)d_in[18];
    const float* cbt = (const float*)d_in[19];
    const float* cwp = (const float*)d_in[20];
    const float* cbp = (const float*)d_in[21];
    const float* fwt = (const float*)d_in[22];
    const float* fbt = (const float*)d_in[23];
    const float* fwp = (const float*)d_in[24];
    const float* fbp = (const float*)d_in[25];
    const int* gsrc = (const int*)d_in[26];
    const int* gdst = (const int*)d_in[27];
    const int* vsrc = (const int*)d_in[28];
    const int* vdst = (const int*)d_in[29];
    const int EG = in_sizes[26];
    const int EV = in_sizes[28];

    // bump allocator over workspace
    char* base = (char*)d_ws;
    size_t off = 0;
    auto alloc = [&](size_t bytes) -> void* {
        void* p = base + off;
        off += (bytes + 255) & ~(size_t)255;
        return p;
    };
    float* fA  = (float*)alloc((size_t)NAP * 64 * 4);
    float* fB  = (float*)alloc((size_t)NAP * 64 * 4);
    float* fV0 = (float*)alloc((size_t)N_VOX * 64 * 4);
    float* fV1 = (float*)alloc((size_t)N_VOX * 64 * 4);
    float* Y   = (float*)alloc((size_t)NAP * 64 * 4);
    float* P   = (float*)alloc((size_t)NAP * 64 * 4);
    int* cntA  = (int*)alloc((size_t)NAP * 4);
    int* curA  = (int*)alloc((size_t)NAP * 4);
    int* roffA = (int*)alloc((size_t)(NAP + 16) * 4);
    int* rsrcA = (int*)alloc((size_t)EG * 4);
    int* cntV  = (int*)alloc((size_t)N_VOX * 4);
    int* curV  = (int*)alloc((size_t)N_VOX * 4);
    int* roffV = (int*)alloc((size_t)(N_VOX + 16) * 4);
    int* rsrcV = (int*)alloc((size_t)EV * 4);

    dim3 B(256);
    auto g1 = [](int n)  { return dim3((unsigned)((n + 255) / 256)); };
    auto gw = [](int nw) { return dim3((unsigned)((nw + 7) / 8)); };  // one wave32 per item

    // CSR (full graph)
    zero_i32<<<g1(NAP), B, 0, stream>>>(cntA, NAP);
    zero_i32<<<g1(NAP), B, 0, stream>>>(curA, NAP);
    count_edges<<<g1(EG), B, 0, stream>>>(gdst, EG, cntA);
    scan_counts<<<dim3(1), B, 0, stream>>>(cntA, N_ALL, EG, roffA);
    fill_csr<<<g1(EG), B, 0, stream>>>(gsrc, gdst, EG, roffA, curA, rsrcA);
    // CSR (voxel graph)
    zero_i32<<<g1(N_VOX), B, 0, stream>>>(cntV, N_VOX);
    zero_i32<<<g1(N_VOX), B, 0, stream>>>(curV, N_VOX);
    count_edges<<<g1(EV), B, 0, stream>>>(vdst, EV, cntV);
    scan_counts<<<dim3(1), B, 0, stream>>>(cntV, N_VOX, EV, roffV);
    fill_csr<<<g1(EV), B, 0, stream>>>(vsrc, vdst, EV, roffV, curV, rsrcV);

    build_x<<<g1(NAP * 64), B, 0, stream>>>(inputs, koor, fA);

    const int tilesA = NAP / 16;    // 538
    const int tilesV = N_VOX / 16;  // 500

    // b1..b8 on full graph (54->54); b8 keeps voxel rows only + dropout u1
    for (int i = 0; i < 8; ++i) {
        gemm_yp<64, 64><<<gw(tilesA), B, 0, stream>>>(fA, bwt + (size_t)i * 54 * 54,
                                                      bwp + (size_t)i * 54 * 54,
                                                      54, 54, tilesA, Y, P);
        if (i < 7) {
            edge_fin<64><<<gw(N_ALL), B, 0, stream>>>(Y, P, roffA, rsrcA,
                                                      bbt + i * 54, bbp + i * 54,
                                                      54, N_ALL, 0, nullptr, fB, 0);
            float* t = fA; fA = fB; fB = t;
        } else {
            edge_fin<64><<<gw(N_VOX), B, 0, stream>>>(Y, P, roffA, rsrcA,
                                                      bbt + 7 * 54, bbp + 7 * 54,
                                                      54, N_VOX, N_PMT, u1, fV0, 0);
        }
    }

    // a1..a4 (54->54) on voxel graph; dropout u2 after a3
    for (int i = 0; i < 4; ++i) {
        gemm_yp<64, 64><<<gw(tilesV), B, 0, stream>>>(fV0, awt + (size_t)i * 54 * 54,
                                                      awp + (size_t)i * 54 * 54,
                                                      54, 54, tilesV, Y, P);
        const float* u = (i == 2) ? u2 : nullptr;
        edge_fin<64><<<gw(N_VOX), B, 0, stream>>>(Y, P, roffV, rsrcV,
                                                  abt + i * 54, abp + i * 54,
                                                  54, N_VOX, 0, u, fV1, 0);
        float* t = fV0; fV0 = fV1; fV1 = t;
    }

    // a5 (54->25)
    gemm_yp<64, 32><<<gw(tilesV), B, 0, stream>>>(fV0, a5wt, a5wp, 54, 25, tilesV, Y, P);
    edge_fin<32><<<gw(N_VOX), B, 0, stream>>>(Y, P, roffV, rsrcV, a5bt, a5bp,
                                              25, N_VOX, 0, nullptr, fV1, 0);
    { float* t = fV0; fV0 = fV1; fV1 = t; }

    // a6..a14 (25->25); dropout u3 after a6 (c[0]), u4 after a9 (c[3])
    for (int i = 0; i < 9; ++i) {
        gemm_yp<32, 32><<<gw(tilesV), B, 0, stream>>>(fV0, cwt + (size_t)i * 25 * 25,
                                                      cwp + (size_t)i * 25 * 25,
                                                      25, 25, tilesV, Y, P);
        const float* u = (i == 0) ? u3 : ((i == 3) ? u4 : nullptr);
        edge_fin<32><<<gw(N_VOX), B, 0, stream>>>(Y, P, roffV, rsrcV,
                                                  cbt + i * 25, cbp + i * 25,
                                                  25, N_VOX, 0, u, fV1, 0);
        float* t = fV0; fV0 = fV1; fV1 = t;
    }

    // a15 (25->1) + ReLU -> d_out[8000]
    gemm_yp<32, 16><<<gw(tilesV), B, 0, stream>>>(fV0, fwt, fwp, 25, 1, tilesV, Y, P);
    edge_fin<16><<<gw(N_VOX), B, 0, stream>>>(Y, P, roffV, rsrcV, fbt, fbp,
                                              1, N_VOX, 0, nullptr, (float*)d_out, 1);
}